// SelfAttentionWithPagedCache_46291157516391
// MI455X (gfx1250) — compile-verified
//
#include <hip/hip_runtime.h>
#include <hip/hip_bf16.h>
#include <math.h>

typedef float v2f __attribute__((ext_vector_type(2)));
typedef float v8f __attribute__((ext_vector_type(8)));
typedef int   i32x4 __attribute__((vector_size(16)));

#define D_MODEL 2048
#define NHEAD   16
#define HD      128
#define BQ      8
#define SNEW    16
#define MROWS   128          // B*S_NEW
#define PAST    4096
#define NTILES  257          // (PAST + SNEW) / 16
#define WAVES   8
#define WBYTES  32768        // per-wave LDS staging block (2x8KB K + 2x8KB V)

#define AS1 __attribute__((address_space(1)))
#define AS3 __attribute__((address_space(3)))

// ---------------------------------------------------------------------------
// CDNA5 async global->LDS copy (ASYNCcnt path), builtin with asm fallback.
// ---------------------------------------------------------------------------
__device__ __forceinline__ void async_ld_b128(const void* gaddr, void* laddr)
{
#if __has_builtin(__builtin_amdgcn_global_load_async_to_lds_b128)
    __builtin_amdgcn_global_load_async_to_lds_b128(
        (AS1 i32x4*)gaddr, (AS3 i32x4*)laddr, 0, 0);
#else
    unsigned lofs = (unsigned)(unsigned long long)(AS3 void*)laddr;
    asm volatile("global_load_async_to_lds_b128 %0, %1, off"
                 :: "v"(lofs), "v"(gaddr) : "memory");
#endif
}

template <int N>
__device__ __forceinline__ void wait_asynccnt()
{
#if __has_builtin(__builtin_amdgcn_s_wait_asynccnt)
    __builtin_amdgcn_s_wait_asynccnt(N);
#else
    if constexpr (N == 0)
        asm volatile("s_wait_asynccnt 0x0" ::: "memory");
    else
        asm volatile("s_wait_asynccnt 0x20" ::: "memory");
#endif
    asm volatile("" ::: "memory");   // keep LDS reads below the wait
}

// ---------------------------------------------------------------------------
// C[M=128, N=2048] = X[128,2048] @ W[2048,2048] + bias
// grid.x = N/16 blocks, 8 waves/block, wave w handles M-tile w.
// f32 WMMA 16x16x4, K stepped by 4.
// ---------------------------------------------------------------------------
__global__ __launch_bounds__(256) void gemm_bias_f32(
    const float* __restrict__ X,
    const float* __restrict__ W0, const float* __restrict__ B0, float* __restrict__ C0,
    const float* __restrict__ W1, const float* __restrict__ B1, float* __restrict__ C1,
    const float* __restrict__ W2, const float* __restrict__ B2, float* __restrict__ C2)
{
    const int lane = threadIdx.x & 31;
    const int wave = threadIdx.x >> 5;
    const int half = lane >> 4;
    const int l16  = lane & 15;

    const int which = blockIdx.y;
    const float* W  = (which == 0) ? W0 : (which == 1) ? W1 : W2;
    const float* Bi = (which == 0) ? B0 : (which == 1) ? B1 : B2;
    float* C        = (which == 0) ? C0 : (which == 1) ? C1 : C2;

    const int n0 = blockIdx.x * 16;
    const int m0 = wave * 16;

    const float* xrow = X + (size_t)(m0 + l16) * D_MODEL + 2 * half;
    const float* wcol = W + n0 + l16;

    v8f acc = {};
    #pragma unroll 8
    for (int k = 0; k < D_MODEL; k += 4) {
        v2f a = *(const v2f*)(xrow + k);
        const int kr = k + 2 * half;
        v2f b;
        b.x = wcol[(size_t)kr * D_MODEL];
        b.y = wcol[(size_t)(kr + 1) * D_MODEL];
        acc = __builtin_amdgcn_wmma_f32_16x16x4_f32(
            false, a, false, b, (short)0, acc, false, false);
    }

    const float bv = Bi[n0 + l16];
    #pragma unroll
    for (int r = 0; r < 8; ++r)
        C[(size_t)(m0 + r + 8 * half) * D_MODEL + n0 + l16] = acc[r] + bv;
}

// ---------------------------------------------------------------------------
// Flash-decode attention with double-buffered async K/V staging.
// grid.x = B*H = 128. 8 waves split the 257 key tiles.
// LDS tile layout: row i (512B) holds its 16B chunks XOR-swizzled: the data of
// linear chunk c is stored at chunk position (c ^ i). Reads use the same map.
// ---------------------------------------------------------------------------
__global__ __launch_bounds__(256) void attn_flash_f32(
    const float* __restrict__ Q,     // [128, 2048]  row = b*16+s, col = h*128+hd
    const float* __restrict__ Kn,    // [128, 2048]  new keys
    const float* __restrict__ Vn,    // [128, 2048]  new values
    const float* __restrict__ Kc,    // [B,H,4096,128]
    const float* __restrict__ Vc,    // [B,H,4096,128]
    float* __restrict__ Oa)          // [128, 2048]
{
    extern __shared__ char lds[];

    const int lane = threadIdx.x & 31;
    const int wave = threadIdx.x >> 5;
    const int half = lane >> 4;
    const int l16  = lane & 15;

    const int bh = blockIdx.x;               // b*16 + h  (H == 16)
    const int b  = bh >> 4;
    const int h  = bh & 15;

    char* wblk = lds + wave * WBYTES;
    char* kbuf[2] = { wblk,          wblk + 8192  };
    char* vbuf[2] = { wblk + 16384,  wblk + 24576 };

    // Preload Q^T B-fragments (32 k-steps) into registers.
    const float* qrow = Q + (size_t)(b * SNEW + l16) * D_MODEL + h * HD + 2 * half;
    v2f qf[32];
    #pragma unroll
    for (int kk = 0; kk < 32; ++kk)
        qf[kk] = *(const v2f*)(qrow + 4 * kk);

    const float* kcb = Kc + (size_t)bh * PAST * HD;
    const float* vcb = Vc + (size_t)bh * PAST * HD;
    const char*  knb = (const char*)(Kn + (size_t)(b * SNEW) * D_MODEL + h * HD);
    const char*  vnb = (const char*)(Vn + (size_t)(b * SNEW) * D_MODEL + h * HD);

    // Issue one tile's K+V (32 x B128 async copies, 512B per instruction).
    auto stage = [&](int t, char* kd, char* vd) {
        const char*  gk = (t < 256) ? (const char*)(kcb + (size_t)t * 16 * HD) : knb;
        const char*  gv = (t < 256) ? (const char*)(vcb + (size_t)t * 16 * HD) : vnb;
        const size_t st = (t < 256) ? 512u : (size_t)D_MODEL * 4;
        #pragma unroll
        for (int i = 0; i < 16; ++i)
            async_ld_b128(gk + (size_t)i * st + lane * 16,
                          kd + i * 512 + ((lane ^ i) << 4));
        #pragma unroll
        for (int i = 0; i < 16; ++i)
            async_ld_b128(gv + (size_t)i * st + lane * 16,
                          vd + i * 512 + ((lane ^ i) << 4));
    };

    v8f o[8];
    #pragma unroll
    for (int c = 0; c < 8; ++c) o[c] = (v8f){};
    float m = -INFINITY;
    float l = 0.0f;
    const float scale = 0.08838834764831845f;   // 1/sqrt(128)

    int cur = 0;
    stage(wave, kbuf[0], vbuf[0]);              // prologue

    for (int t = wave; t < NTILES; t += WAVES) {
        const int tn = t + WAVES;
        if (tn < NTILES) {
            stage(tn, kbuf[cur ^ 1], vbuf[cur ^ 1]);
            wait_asynccnt<32>();                // current tile's 32 copies done
        } else {
            wait_asynccnt<0>();
        }

        // --- scores: S^T(16 keys x 16 queries) = K_tile * Q^T --------------
        const char* ka = kbuf[cur] + l16 * 512 + 8 * half;
        v8f s = {};
        #pragma unroll
        for (int kk = 0; kk < 32; ++kk) {
            v2f a = *(const v2f*)(ka + ((kk ^ l16) << 4));
            s = __builtin_amdgcn_wmma_f32_16x16x4_f32(
                false, a, false, qf[kk], (short)0, s, false, false);
        }

        // --- scale + causal mask (final tile holds the 16 new keys) --------
        float sv[8];
        #pragma unroll
        for (int r = 0; r < 8; ++r) {
            float x = s[r] * scale;
            if (t == 256) {
                const int kl = r + 8 * half;    // key index within new block
                if (kl > l16) x = -INFINITY;    // query = l16
            }
            sv[r] = x;
        }

        // --- online softmax (stats per query = lane%16) --------------------
        float rmax = sv[0];
        #pragma unroll
        for (int r = 1; r < 8; ++r) rmax = fmaxf(rmax, sv[r]);
        rmax = fmaxf(rmax, __shfl_xor(rmax, 16));
        const float mnew = fmaxf(m, rmax);

        float psum = 0.0f;
        #pragma unroll
        for (int r = 0; r < 8; ++r) {
            sv[r] = __expf(sv[r] - mnew);
            psum += sv[r];
        }
        psum += __shfl_xor(psum, 16);
        const float alpha = __expf(m - mnew);
        l = l * alpha + psum;
        m = mnew;

        float ar[8];
        #pragma unroll
        for (int r = 0; r < 8; ++r) ar[r] = __shfl(alpha, r + 8 * half);
        #pragma unroll
        for (int c = 0; c < 8; ++c)
            #pragma unroll
            for (int r = 0; r < 8; ++r) o[c][r] *= ar[r];

        // --- O += P * V_tile ----------------------------------------------
        // P^T element (key, query) lives at lane query+16*(key>=8), VGPR key&7.
        #pragma unroll
        for (int kk = 0; kk < 4; ++kk) {
            const int src = l16 + ((kk >= 2) ? 16 : 0);
            const float a0A = __shfl(sv[(4 * kk + 0) & 7], src);
            const float a0B = __shfl(sv[(4 * kk + 2) & 7], src);
            const float a1A = __shfl(sv[(4 * kk + 1) & 7], src);
            const float a1B = __shfl(sv[(4 * kk + 3) & 7], src);
            v2f a;
            a.x = half ? a0B : a0A;
            a.y = half ? a1B : a1A;

            const int kr0 = 4 * kk + 2 * half;
            const int kr1 = kr0 + 1;
            const char* v0 = vbuf[cur] + kr0 * 512 + (l16 & 3) * 4;
            const char* v1 = vbuf[cur] + kr1 * 512 + (l16 & 3) * 4;
            #pragma unroll
            for (int c = 0; c < 8; ++c) {
                const int ch = 4 * c + (l16 >> 2);
                v2f bf;
                bf.x = *(const float*)(v0 + ((ch ^ kr0) << 4));
                bf.y = *(const float*)(v1 + ((ch ^ kr1) << 4));
                o[c] = __builtin_amdgcn_wmma_f32_16x16x4_f32(
                    false, a, false, bf, (short)0, o[c], false, false);
            }
        }
        cur ^= 1;
    }

    // --- merge partials across waves (combine region aliases own staging) --
    float* ow = (float*)wblk;                       // 2048 floats O + 32 stats
    #pragma unroll
    for (int c = 0; c < 8; ++c)
        #pragma unroll
        for (int r = 0; r < 8; ++r)
            ow[c * 256 + r * 32 + lane] = o[c][r];
    if (half == 0) {
        ow[2048 + l16] = m;
        ow[2064 + l16] = l;
    }
    __syncthreads();

    const int c = wave;   // each wave finalizes one 16-wide HD chunk
    #pragma unroll
    for (int r = 0; r < 8; ++r) {
        const int q = r + 8 * half;
        float M = -INFINITY;
        #pragma unroll
        for (int w = 0; w < WAVES; ++w)
            M = fmaxf(M, ((const float*)(lds + w * WBYTES))[2048 + q]);
        float Lt = 0.0f, acc = 0.0f;
        #pragma unroll
        for (int w = 0; w < WAVES; ++w) {
            const float* lw = (const float*)(lds + w * WBYTES);
            const float sc = __expf(lw[2048 + q] - M);
            Lt  += sc * lw[2064 + q];
            acc += sc * lw[c * 256 + r * 32 + lane];
        }
        Oa[(size_t)(b * SNEW + q) * D_MODEL + h * HD + 16 * c + l16] = acc / Lt;
    }
}

// ---------------------------------------------------------------------------
extern "C" void kernel_launch(void* const* d_in, const int* in_sizes, int n_in,
                              void* d_out, int out_size, void* d_ws, size_t ws_size,
                              hipStream_t stream)
{
    (void)in_sizes; (void)n_in; (void)out_size; (void)ws_size;

    const float* x  = (const float*)d_in[0];
    const float* kc = (const float*)d_in[1];
    const float* vc = (const float*)d_in[2];
    const float* Wq = (const float*)d_in[3];
    const float* bq = (const float*)d_in[4];
    const float* Wk = (const float*)d_in[5];
    const float* bk = (const float*)d_in[6];
    const float* Wv = (const float*)d_in[7];
    const float* bv = (const float*)d_in[8];
    const float* Wo = (const float*)d_in[9];
    const float* bo = (const float*)d_in[10];
    float* out = (float*)d_out;

    float* Qw = (float*)d_ws;                 // [128,2048]
    float* Kn = Qw + (size_t)MROWS * D_MODEL;
    float* Vn = Kn + (size_t)MROWS * D_MODEL;
    float* Ao = Vn + (size_t)MROWS * D_MODEL;

    gemm_bias_f32<<<dim3(D_MODEL / 16, 3), 256, 0, stream>>>(
        x, Wq, bq, Qw, Wk, bk, Kn, Wv, bv, Vn);

    attn_flash_f32<<<dim3(BQ * NHEAD), 256, WAVES * WBYTES, stream>>>(
        Qw, Kn, Vn, kc, vc, Ao);

    gemm_bias_f32<<<dim3(D_MODEL / 16, 1), 256, 0, stream>>>(
        Ao, Wo, bo, out, Wo, bo, out, Wo, bo, out);
}